// GFSDTWLayer_36910948942102
// MI455X (gfx1250) — compile-verified
//
#include <hip/hip_runtime.h>
#include <hip/hip_bf16.h>

// ---------------------------------------------------------------------------
// Multi-level soft-DTW (GFSDTW) for MI455X / gfx1250, wave32.
//
// Pass 1: pooling of x[4096,256] to all 8 pyramid levels as a single matmul
//         XL = X @ P  using V_WMMA_F32_16X16X4_F32 (exact f32 accumulate).
//         P[256,510] is the block-averaging matrix (levels concatenated,
//         level offset = 2^(l+1)-2), generated on the fly per K-chunk.
// Pass 2: soft-DTW DP, one wave per (level, batch) element, anti-diagonal
//         systolic wavefront across 32 lanes (32 DP rows per band),
//         up/diag passed lane-to-lane via __shfl_up, band bottom row handed
//         to the next band through a per-wave LDS row buffer (same-wave LDS
//         ops are architecturally in-order; no barrier needed inside a wave).
// ---------------------------------------------------------------------------

typedef float v2f __attribute__((ext_vector_type(2)));
typedef float v8f __attribute__((ext_vector_type(8)));

#define BIGV   1e9f
#define GAMMA_C 0.1f
#define INVG   10.0f
#define NBATCH 4096
#define TCOLS  510   // 2+4+8+16+32+64+128+256

// ------------------------------ Pass 1: WMMA pooling ------------------------
// One wave computes one 16x16 tile of XL[4096 x 510].
// A tile (16x4 f32): lanes 0-15 hold M=lane, K={kc,kc+1}; lanes 16-31 hold
// M=lane-16, K={kc+2,kc+3}  (per CDNA5 ISA 32-bit A-matrix layout).
// B tile (4x16 f32) assumed mirrored: lanes 0-15 K={kc,kc+1}, lanes 16-31
// K={kc+2,kc+3}, N = lane&15.
// D/C (16x16 f32): VGPR v holds M=v (lanes 0-15) / M=v+8 (lanes 16-31), N=lane&15.
__global__ __launch_bounds__(256)
void pool_wmma_kernel(const float* __restrict__ x, float* __restrict__ xl_all) {
    const int lane = threadIdx.x & 31;
    const int wid  = threadIdx.x >> 5;
    const int tile = blockIdx.x * 8 + wid;      // 8192 tiles total
    const int rt   = tile >> 5;                 // 256 row tiles (batch/16)
    const int ct   = tile & 31;                 // 32 col tiles (ceil(510/16))

    const int r0   = rt << 4;
    const int gcol = (ct << 4) + (lane & 15);
    const bool colvalid = (gcol < TCOLS);

    // Decode (level, j, pooling width) from the concatenated column index.
    // level offsets are 2^(l+1)-2, so level = floor(log2(gcol+2)) - 1.
    int   level = 0, j = 0, shift = 0;
    float wsc = 0.0f;
    if (colvalid) {
        level = 30 - __clz(gcol + 2);           // 0..7
        j     = gcol - ((2 << level) - 2);      // column within level
        shift = 7 - level;                      // pooling block = 1<<shift
        wsc   = 1.0f / (float)(1 << shift);     // exact (power of two)
    }

    const int   klo  = (lane >> 4) * 2;         // lane-half K offset
    const float* xrow = x + (size_t)(r0 + (lane & 15)) * 256;

    v8f acc = {};
    for (int kc = 0; kc < 256; kc += 4) {
        v2f a;
        a.x = xrow[kc + klo];
        a.y = xrow[kc + klo + 1];
        const int kk = kc + klo;
        v2f b;
        b.x = (colvalid && ((kk       >> shift) == j)) ? wsc : 0.0f;
        b.y = (colvalid && (((kk + 1) >> shift) == j)) ? wsc : 0.0f;
        acc = __builtin_amdgcn_wmma_f32_16x16x4_f32(
                  false, a, false, b, (short)0, acc, false, false);
    }

    if (colvalid) {
        const int mbase = (lane >> 4) * 8;
        #pragma unroll
        for (int v = 0; v < 8; ++v)
            xl_all[(size_t)(r0 + mbase + v) * TCOLS + gcol] = acc[v];
    }
}

// ------------------------------ Pass 2: wavefront DP ------------------------
struct KPtrs { const float* k[8]; };

__device__ __forceinline__ float softmin3(float a, float b, float c) {
    const float m = fminf(a, fminf(b, c));
    const float e = __expf((m - a) * INVG)
                  + __expf((m - b) * INVG)
                  + __expf((m - c) * INVG);
    return m - GAMMA_C * __logf(e);
}

__global__ __launch_bounds__(256)
void sdtw_dp_kernel(const float* __restrict__ xl_all, KPtrs kp,
                    float* __restrict__ out) {
    __shared__ float s_xl [8][256];   // pooled series, one slab per wave
    __shared__ float s_row[8][256];   // band-boundary DP row, one per wave

    const int lane  = threadIdx.x & 31;
    const int wid   = threadIdx.x >> 5;
    const int level = blockIdx.y;                   // 0..7
    const int b     = blockIdx.x * 8 + wid;         // batch element
    const int L     = 2 << level;                   // 2..256
    const int off   = (2 << level) - 2;             // column offset in xl_all

    float* __restrict__ xl  = s_xl[wid];
    float* __restrict__ row = s_row[wid];
    const float* __restrict__ kptr = kp.k[level];
    const float* __restrict__ src  = xl_all + (size_t)b * TCOLS + off;

    for (int c = lane; c < L; c += 32) {
        xl[c]  = src[c];
        row[c] = BIGV;                 // R[-1][j] boundary for band 0
    }
    __syncthreads();                   // one-time; all waves take it once

    const int nbands = (L + 31) >> 5;
    float cur = BIGV;

    for (int band = 0; band < nbands; ++band) {
        const int  i0     = band << 5;
        const int  br     = min(32, L - i0);
        const bool rvalid = (lane < br);
        const float kv    = rvalid ? kptr[i0 + lane] : 0.0f;

        cur = BIGV;
        // uprev doubles as diag source; corner R[-1][-1]=0 only at origin.
        float uprev = (band == 0 && lane == 0) ? 0.0f : BIGV;

        const int steps = L + br - 1;
        for (int t = 0; t < steps; ++t) {
            // up = neighbor lane's value from the previous step (R[i-1][j])
            float u = __shfl_up(cur, 1, 32);
            const int j = t - lane;
            if (lane == 0) {
                // row above this band: previous band's bottom row (or BIG)
                u = (band != 0 && j < L) ? row[j] : BIGV;
            }
            const bool active = rvalid && (j >= 0) && (j < L);

            const float xv   = xl[active ? j : 0];
            const float dd   = xv - kv;
            const float left = (j == 0) ? BIGV : cur;
            const float val  = dd * dd + softmin3(left, u, uprev);

            if (active) {
                cur = val;
                if (lane == br - 1) row[j] = val;  // hand to next band
            }
            uprev = u;                              // becomes diag next step
        }
    }

    // R[L-1][L-1] lives in lane (L-1) & 31 after the last band.
    if (lane == ((L - 1) & 31))
        out[level * NBATCH + b] = cur;
}

// ------------------------------ launcher ------------------------------------
extern "C" void kernel_launch(void* const* d_in, const int* in_sizes, int n_in,
                              void* d_out, int out_size, void* d_ws, size_t ws_size,
                              hipStream_t stream) {
    const float* x = (const float*)d_in[0];
    KPtrs kp;
    for (int i = 0; i < 8; ++i) kp.k[i] = (const float*)d_in[1 + i];

    float* xl_all = (float*)d_ws;          // 4096 * 510 * 4B = ~8.36 MB

    // Pass 1: 8192 tiles, 8 waves/block -> 1024 blocks.
    pool_wmma_kernel<<<1024, 256, 0, stream>>>(x, xl_all);

    // Pass 2: one wave per (level, batch): grid (4096/8, 8 levels).
    dim3 grid(NBATCH / 8, 8);
    sdtw_dp_kernel<<<grid, 256, 0, stream>>>(xl_all, kp, (float*)d_out);
}